// SUPREME_1932735284046
// MI455X (gfx1250) — compile-verified
//
#include <hip/hip_runtime.h>

typedef __attribute__((ext_vector_type(2))) float v2f;
typedef __attribute__((ext_vector_type(8))) float v8f;

#define N_NODES  100000
#define N_EDGES  1600000
#define DIM_IN   256
#define DIM_HID  128
#define DIM_OUT  32

// ---------------- degree / normalization ----------------

__global__ __launch_bounds__(256) void k_init_deg(float* deg, int n) {
  int i = blockIdx.x * blockDim.x + threadIdx.x;
  if (i < n) deg[i] = 1.0f;  // self-loop contributes weight 1
}

__global__ __launch_bounds__(256) void k_deg_accum(const int* __restrict__ dst,
                                                   const float* __restrict__ w,
                                                   float* deg, int e) {
  int i = blockIdx.x * blockDim.x + threadIdx.x;
  if (i < e) atomicAdd(&deg[dst[i]], w[i]);
}

__global__ __launch_bounds__(256) void k_deg_to_inv(float* deg, int n) {
  int i = blockIdx.x * blockDim.x + threadIdx.x;
  if (i < n) {
    float d = deg[i];
    deg[i] = (d > 0.0f) ? rsqrtf(fmaxf(d, 1e-12f)) : 0.0f;
  }
}

__global__ __launch_bounds__(256) void k_init_bias(float* buf,
                                                   const float* __restrict__ bias,
                                                   int mask, long total) {
  long i = (long)blockIdx.x * blockDim.x + threadIdx.x;
  if (i < total) buf[i] = bias[i & mask];
}

// ---------------- GEMM1: h1 = x @ W1  (fp32 WMMA 16x16x4) ----------------
// One wave computes one 16x16 tile of H1. 6250 row-tiles x 8 col-tiles.
// A-frag (16x4 f32, ISA 7.12.2): lanes 0-15 -> M=lane, K={0,1}; lanes 16-31 -> K={2,3}.
// B-frag (4x16 f32): lanes 0-15 -> N=lane, K rows {0,1}; lanes 16-31 -> K rows {2,3}.
// C/D (16x16 f32): vgpr i: lanes 0-15 -> M=i, N=lane; lanes 16-31 -> M=8+i, N=lane-16.

__global__ __launch_bounds__(256) void k_gemm1_wmma(const float* __restrict__ X,
                                                    const float* __restrict__ W,
                                                    float* __restrict__ H) {
  const int gw   = blockIdx.x * 8 + (threadIdx.x >> 5);  // global wave id
  const int lane = threadIdx.x & 31;
  const int rt = gw >> 3;          // row tile (0..6249)
  const int ct = gw & 7;           // col tile (0..7)
  const int m0 = rt * 16, n0 = ct * 16;
  const int l    = lane & 15;
  const int half = lane >> 4;      // 0 or 1
  const int ko   = half * 2;       // K sub-offset {0,2}

  const float* arow = X + (size_t)(m0 + l) * DIM_IN;
  v8f c = {0.f, 0.f, 0.f, 0.f, 0.f, 0.f, 0.f, 0.f};

  for (int k0 = 0; k0 < DIM_IN; k0 += 4) {
    v2f a = *(const v2f*)(arow + k0 + ko);              // x[m][k0+ko], x[m][k0+ko+1]
    v2f b;
    b.x = W[(size_t)(k0 + ko)     * DIM_HID + n0 + l];  // W[k][n]
    b.y = W[(size_t)(k0 + ko + 1) * DIM_HID + n0 + l];
    c = __builtin_amdgcn_wmma_f32_16x16x4_f32(false, a, false, b,
                                              (short)0, c, false, false);
  }

  float* drow = H + (size_t)(m0 + half * 8) * DIM_HID + (n0 + l);
#pragma unroll
  for (int i = 0; i < 8; ++i) drow[(size_t)i * DIM_HID] = c[i];
}

// ---------------- GEMM2: h2 = relu(x_emb) @ W2 (fp32 WMMA) ----------------

__global__ __launch_bounds__(256) void k_gemm2_wmma(const float* __restrict__ XE,
                                                    const float* __restrict__ W,
                                                    float* __restrict__ H) {
  const int gw = blockIdx.x * 8 + (threadIdx.x >> 5);
  if (gw >= (N_NODES / 16) * (DIM_OUT / 16)) return;   // wave-uniform guard
  const int lane = threadIdx.x & 31;
  const int rt = gw >> 1;          // row tile (0..6249)
  const int ct = gw & 1;           // col tile (0..1)
  const int m0 = rt * 16, n0 = ct * 16;
  const int l    = lane & 15;
  const int half = lane >> 4;
  const int ko   = half * 2;

  const float* arow = XE + (size_t)(m0 + l) * DIM_HID;
  v8f c = {0.f, 0.f, 0.f, 0.f, 0.f, 0.f, 0.f, 0.f};

  for (int k0 = 0; k0 < DIM_HID; k0 += 4) {
    v2f a;
    a.x = fmaxf(arow[k0 + ko],     0.0f);               // fused ReLU
    a.y = fmaxf(arow[k0 + ko + 1], 0.0f);
    v2f b;
    b.x = W[(size_t)(k0 + ko)     * DIM_OUT + n0 + l];
    b.y = W[(size_t)(k0 + ko + 1) * DIM_OUT + n0 + l];
    c = __builtin_amdgcn_wmma_f32_16x16x4_f32(false, a, false, b,
                                              (short)0, c, false, false);
  }

  float* drow = H + (size_t)(m0 + half * 8) * DIM_OUT + (n0 + l);
#pragma unroll
  for (int i = 0; i < 8; ++i) drow[(size_t)i * DIM_OUT] = c[i];
}

// ---------------- edge scatter (message passing) ----------------
// One wave per edge (including N synthetic self-loop edges at the tail).

__global__ __launch_bounds__(256) void k_scatter128(const int* __restrict__ src,
                                                    const int* __restrict__ dst,
                                                    const float* __restrict__ wgt,
                                                    const float* __restrict__ inv,
                                                    const float* __restrict__ h,
                                                    float* acc, int e, int n) {
  const int wid  = blockIdx.x * 8 + (threadIdx.x >> 5);
  const int lane = threadIdx.x & 31;
  if (wid >= e + n) return;
  int s, t; float w;
  if (wid < e) { s = src[wid]; t = dst[wid]; w = wgt[wid]; }
  else         { s = t = wid - e; w = 1.0f; }
  const float nrm = inv[s] * w * inv[t];
  const float4 v = *(const float4*)(h + (size_t)s * DIM_HID + lane * 4);
  float* o = acc + (size_t)t * DIM_HID + lane * 4;
  atomicAdd(o + 0, nrm * v.x);
  atomicAdd(o + 1, nrm * v.y);
  atomicAdd(o + 2, nrm * v.z);
  atomicAdd(o + 3, nrm * v.w);
}

__global__ __launch_bounds__(256) void k_scatter32(const int* __restrict__ src,
                                                   const int* __restrict__ dst,
                                                   const float* __restrict__ wgt,
                                                   const float* __restrict__ inv,
                                                   const float* __restrict__ h,
                                                   float* acc, int e, int n) {
  const int wid  = blockIdx.x * 8 + (threadIdx.x >> 5);
  const int lane = threadIdx.x & 31;
  if (wid >= e + n) return;
  int s, t; float w;
  if (wid < e) { s = src[wid]; t = dst[wid]; w = wgt[wid]; }
  else         { s = t = wid - e; w = 1.0f; }
  const float nrm = inv[s] * w * inv[t];
  const float v = h[(size_t)s * DIM_OUT + lane];
  atomicAdd(acc + (size_t)t * DIM_OUT + lane, nrm * v);
}

// ---------------- driver ----------------

extern "C" void kernel_launch(void* const* d_in, const int* in_sizes, int n_in,
                              void* d_out, int out_size, void* d_ws, size_t ws_size,
                              hipStream_t stream) {
  const int N = N_NODES, E = N_EDGES;

  const float* x   = (const float*)d_in[0];
  const int*   ei  = (const int*)d_in[1];     // [2, E] flattened
  const int*   src = ei;
  const int*   dst = ei + E;
  const float* w   = (const float*)d_in[2];
  const float* W1  = (const float*)d_in[3];
  const float* b1  = (const float*)d_in[4];
  const float* W2  = (const float*)d_in[5];
  const float* b2  = (const float*)d_in[6];

  float* out  = (float*)d_out;                      // [N, 32]
  float* xemb = out + (size_t)N * DIM_OUT;          // [N, 128]

  float* inv = (float*)d_ws;                        // N floats (deg -> inv_sqrt)
  float* h1  = inv + N;                             // N*128 floats
  float* h2  = h1 + (size_t)N * DIM_HID;            // N*32 floats

  // 1. symmetric GCN normalization
  k_init_deg  <<<(N + 255) / 256, 256, 0, stream>>>(inv, N);
  k_deg_accum <<<(E + 255) / 256, 256, 0, stream>>>(dst, w, inv, E);
  k_deg_to_inv<<<(N + 255) / 256, 256, 0, stream>>>(inv, N);

  // 2. h1 = x @ W1   (6250 row-tiles * 8 col-tiles, 8 waves/block)
  k_gemm1_wmma<<<(N / 16) * (DIM_HID / 16) / 8, 256, 0, stream>>>(x, W1, h1);

  // 3. x_emb = segsum(norm * h1[src]) + b1
  const long totH = (long)N * DIM_HID;
  k_init_bias <<<(int)((totH + 255) / 256), 256, 0, stream>>>(xemb, b1, DIM_HID - 1, totH);
  k_scatter128<<<(E + N) / 8, 256, 0, stream>>>(src, dst, w, inv, h1, xemb, E, N);

  // 4. h2 = relu(x_emb) @ W2
  const int tiles2 = (N / 16) * (DIM_OUT / 16);
  k_gemm2_wmma<<<(tiles2 + 7) / 8, 256, 0, stream>>>(xemb, W2, h2);

  // 5. out = segsum(norm * h2[src]) + b2
  const long totO = (long)N * DIM_OUT;
  k_init_bias<<<(int)((totO + 255) / 256), 256, 0, stream>>>(out, b2, DIM_OUT - 1, totO);
  k_scatter32<<<(E + N) / 8, 256, 0, stream>>>(src, dst, w, inv, h2, out, E, N);
}